// REDRESS_49374944035230
// MI455X (gfx1250) — compile-verified
//
#include <hip/hip_runtime.h>
#include <math.h>

#define F_MAX_NUM 2000000.0f
#define F_SIGMA_1 0.02f
#define MAXL      64      // supports L = k_para*top_k up to 64
#define BLOCK     256

// ---- CDNA5 async global<->LDS DMA helpers (ASYNCcnt-tracked) --------------
// Builtin signature (from hipcc diagnostic): first param is
// `int __attribute__((vector_size(16))) __device__ *` = v4i addrspace(1)*.
typedef int v4i __attribute__((vector_size(16)));
typedef __attribute__((address_space(1))) v4i v4i_glb;
typedef __attribute__((address_space(3))) v4i v4i_lds;

__device__ __forceinline__ void async_ld16(const float* g, float* l) {
#if __has_builtin(__builtin_amdgcn_global_load_async_to_lds_b128)
  __builtin_amdgcn_global_load_async_to_lds_b128((v4i_glb*)g, (v4i_lds*)l, 0,
                                                 0);
#else
  *(float4*)l = *(const float4*)g;
#endif
}

__device__ __forceinline__ void async_st16(float* g, const float* l) {
#if __has_builtin(__builtin_amdgcn_global_store_async_from_lds_b128)
  __builtin_amdgcn_global_store_async_from_lds_b128((v4i_glb*)g, (v4i_lds*)l,
                                                    0, 0);
#else
  *(float4*)g = *(const float4*)l;
#endif
}

__device__ __forceinline__ void wait_async0() {
#if __has_builtin(__builtin_amdgcn_s_wait_asynccnt)
  __builtin_amdgcn_s_wait_asynccnt(0);
#else
  asm volatile("s_wait_asynccnt 0" ::: "memory");
#endif
}

// ---- block-wide iterative top-k selection out of LDS ----------------------
// Runs `rounds` rounds of argmax over s[0..n), marking winners with -1.0f
// (all real values are >= 0). Tie-break: smaller index wins (argsort
// stability of the reference). 256 threads, wave32.
__device__ __forceinline__ void select_top(float* s, int n, int rounds,
                                           float* tv, int* ti,
                                           float* w_val, int* w_idx) {
  const int tid  = threadIdx.x;
  const int lane = tid & 31;
  const int wid  = tid >> 5;
  for (int t = 0; t < rounds; ++t) {
    float bv = -2.0f;
    int   bi = -1;
    for (int i = tid; i < n; i += BLOCK) {     // conflict-free strided LDS scan
      float v = s[i];
      if (v > bv || (v == bv && i < bi)) { bv = v; bi = i; }
    }
    // wave32 reduction
    for (int off = 16; off > 0; off >>= 1) {
      float ov = __shfl_down(bv, off);
      int   oi = __shfl_down(bi, off);
      if (ov > bv || (ov == bv && oi < bi)) { bv = ov; bi = oi; }
    }
    if (lane == 0) { w_val[wid] = bv; w_idx[wid] = bi; }
    __syncthreads();
    if (tid == 0) {
      float fv = w_val[0];
      int   fi = w_idx[0];
      for (int w = 1; w < (BLOCK >> 5); ++w) {
        float ov = w_val[w];
        int   oi = w_idx[w];
        if (ov > fv || (ov == fv && oi < fi)) { fv = ov; fi = oi; }
      }
      tv[t] = fv;
      ti[t] = fi;
      s[fi] = -1.0f;  // remove winner for the next round
    }
    __syncthreads();
  }
}

// ---- one workgroup per row ------------------------------------------------
__global__ __launch_bounds__(BLOCK) void redress_row_kernel(
    const float* __restrict__ x_sim, const float* __restrict__ y_sim,
    const int* __restrict__ topk_p, const int* __restrict__ kpara_p,
    float* __restrict__ out, int N) {
  const int row = blockIdx.x;
  const int tid = threadIdx.x;

  const int top_k = topk_p[0];
  int L = kpara_p[0] * top_k;
  if (L > MAXL) L = MAXL;
  const int Lp1 = L + 1;

  extern __shared__ float smem[];   // 2*N floats
  float* s_y = smem;                // y row, later reused as mid-row buffer
  float* s_x = smem + N;            // x row

  __shared__ float w_val[BLOCK / 32];
  __shared__ int   w_idx[BLOCK / 32];
  __shared__ float top_val_y[MAXL + 1];
  __shared__ int   top_idx_y[MAXL + 1];
  __shared__ float top_val_x[MAXL + 1];
  __shared__ int   top_idx_x[MAXL + 1];
  __shared__ float s_xcorr[MAXL];
  __shared__ float s_lambda[MAXL];
  __shared__ float s_idcg;

  // ---- stage both rows into LDS via async DMA (16B per lane per issue) ----
  const float* gx = x_sim + (size_t)row * N;
  const float* gy = y_sim + (size_t)row * N;
  for (int i = tid * 4; i < N; i += BLOCK * 4) {
    async_ld16(gy + i, s_y + i);
    async_ld16(gx + i, s_x + i);
  }
  wait_async0();
  __syncthreads();
  if (tid == 0) { s_y[row] = F_MAX_NUM; s_x[row] = F_MAX_NUM; }
  __syncthreads();

  // ---- top-(L+1) of y row (values + indices); rank 0 is the diagonal ------
  select_top(s_y, N, Lp1, top_val_y, top_idx_y, w_val, w_idx);

  // ---- x_corr gather from the pristine x row (before x selection) ---------
  if (tid < L) s_xcorr[tid] = s_x[top_idx_y[tid + 1]];
  __syncthreads();

  // ---- top-(L+1) values of x row ------------------------------------------
  select_top(s_x, N, Lp1, top_val_x, top_idx_x, w_val, w_idx);

  // ---- idcg ----------------------------------------------------------------
  if (tid == 0) {
    float idcg = 0.0f;
    for (int i = 0; i < top_k; ++i)
      idcg += (exp2f(top_val_x[i + 1]) - 1.0f) / log2f(2.0f + (float)i);
    s_idcg = idcg;
  }
  __syncthreads();

  // ---- L x L lambda computation (thread j handles row j of the pair grid) -
  const float rowmask = ((float)row < 0.6f * (float)N) ? 1.0f : 0.0f;
  if (tid < L) {
    const int   j        = tid;
    const float xj       = s_xcorr[j];
    const float yj       = top_val_y[j + 1];
    const float gj       = exp2f(xj) - 1.0f;
    const float dj       = (j < top_k) ? (1.0f / log2f(2.0f + (float)j)) : 0.0f;
    const float inv_idcg = 1.0f / s_idcg;
    float acc = 0.0f;
    for (int k = 0; k < L; ++k) {
      if (k == j) continue;
      const float xk   = s_xcorr[k];
      const float yk   = top_val_y[k + 1];
      const float gk   = exp2f(xk) - 1.0f;
      const float dk   = (k < top_k) ? (1.0f / log2f(2.0f + (float)k)) : 0.0f;
      const float Sjk  = (xj - xk > 0.0f) ? 1.0f : 0.0f;
      const float Skj  = (xk - xj > 0.0f) ? 1.0f : 0.0f;
      const float pm   = (j < k) ? Sjk : Skj;          // pair_mask (symmetric)
      const float nd   = pm * fabsf((gj - gk) * (dj - dk)) * inv_idcg * rowmask;
      const float fjk  = -F_SIGMA_1 / (1.0f + expf((yj - yk) * F_SIGMA_1));
      const float fkj  = -F_SIGMA_1 / (1.0f + expf((yk - yj) * F_SIGMA_1));
      acc += Sjk * fjk * nd - Skj * fkj * nd;          // wz(j,k) - wz(k,j)
    }
    s_lambda[j] = acc;
  }
  __syncthreads();

  // ---- compose mid row in LDS (reuse s_y): zeros + scattered lambdas ------
  for (int i = tid; i < N; i += BLOCK) s_y[i] = 0.0f;
  __syncthreads();
  if (tid < L) s_y[top_idx_y[tid + 1]] = s_lambda[tid];
  __syncthreads();

  // ---- outputs: mid | x_ss | y_si | x_corr (flat concatenation) -----------
  float* mid  = out;
  float* oxss = out + (size_t)N * N;
  float* oysi = oxss + (size_t)N * L;
  float* oxc  = oysi + (size_t)N * L;

  float* mrow = mid + (size_t)row * N;
  for (int i = tid * 4; i < N; i += BLOCK * 4) async_st16(mrow + i, s_y + i);

  if (tid < L) {
    const size_t o = (size_t)row * L + tid;
    oxss[o] = top_val_x[tid + 1];
    oysi[o] = (float)top_idx_y[tid + 1];   // int output stored numerically
    oxc[o]  = s_xcorr[tid];
  }
  wait_async0();
}

extern "C" void kernel_launch(void* const* d_in, const int* in_sizes, int n_in,
                              void* d_out, int out_size, void* d_ws,
                              size_t ws_size, hipStream_t stream) {
  (void)n_in; (void)out_size; (void)d_ws; (void)ws_size;
  const float* x_sim = (const float*)d_in[0];
  const float* y_sim = (const float*)d_in[1];
  const int*   topk  = (const int*)d_in[2];
  const int*   kpara = (const int*)d_in[3];

  const int N = (int)(sqrt((double)in_sizes[0]) + 0.5);  // 4096
  const size_t shmem = (size_t)2 * (size_t)N * sizeof(float);  // 32 KB

  redress_row_kernel<<<N, BLOCK, shmem, stream>>>(x_sim, y_sim, topk, kpara,
                                                  (float*)d_out, N);
}